// LightGCN_xij_item_personal_single_50534585204898
// MI455X (gfx1250) — compile-verified
//
#include <hip/hip_runtime.h>
#include <hip/hip_bf16.h>

#define NUM_USERS 100000
#define NUM_ITEMS 50000
#define N_NODES   (NUM_USERS + NUM_ITEMS)
#define LATENT    64
#define XDIM      16
#define N_LAYERS  3
#define NNZ_CNT   6400000LL
#define BATCH     8192

typedef __attribute__((ext_vector_type(2))) float v2f;
typedef __attribute__((ext_vector_type(8))) float v8f;

// ---------------------------------------------------------------------------
// init: cur = concat(emb_user, emb_item); acc = cur; nxt = 0
// ---------------------------------------------------------------------------
__global__ void lg_init_kernel(const float* __restrict__ eu,
                               const float* __restrict__ ei,
                               float* __restrict__ cur,
                               float* __restrict__ acc,
                               float* __restrict__ nxt) {
    long long i = (long long)blockIdx.x * blockDim.x + threadIdx.x;
    const long long n = (long long)N_NODES * LATENT;
    if (i >= n) return;
    const long long ubound = (long long)NUM_USERS * LATENT;
    float v = (i < ubound) ? eu[i] : ei[i - ubound];
    cur[i] = v;
    acc[i] = v;
    nxt[i] = 0.0f;
}

// ---------------------------------------------------------------------------
// COO SpMM scatter: 16 threads per nnz, float4 per thread, f32 atomics (L2)
// ---------------------------------------------------------------------------
__global__ void __launch_bounds__(256)
lg_spmm_kernel(const int* __restrict__ rows, const int* __restrict__ cols,
               const float* __restrict__ vals, const float* __restrict__ x,
               float* __restrict__ y) {
    long long gid = (long long)blockIdx.x * blockDim.x + threadIdx.x;
    long long e = gid >> 4;
    if (e >= NNZ_CNT) return;
    int q = (int)(gid & 15);              // which float4 of the 64-dim row
    int r = rows[e];
    int c = cols[e];
    float v = vals[e];
    const float4 xv = *(const float4*)(x + (long long)c * LATENT + q * 4);
    float* dst = y + (long long)r * LATENT + q * 4;
    atomicAdd(dst + 0, v * xv.x);
    atomicAdd(dst + 1, v * xv.y);
    atomicAdd(dst + 2, v * xv.z);
    atomicAdd(dst + 3, v * xv.w);
}

// ---------------------------------------------------------------------------
// acc += nxt; zero the other ping-pong buffer for the next layer's output
// (vectorized float4: N_NODES*LATENT divisible by 4)
// ---------------------------------------------------------------------------
__global__ void lg_accum_kernel(float* __restrict__ acc,
                                const float* __restrict__ nxt,
                                float* __restrict__ zero_me) {
    long long i = (long long)blockIdx.x * blockDim.x + threadIdx.x;
    const long long n4 = (long long)N_NODES * LATENT / 4;
    if (i >= n4) return;
    float4 a = ((const float4*)acc)[i];
    float4 b = ((const float4*)nxt)[i];
    a.x += b.x; a.y += b.y; a.z += b.z; a.w += b.w;
    ((float4*)acc)[i] = a;
    ((float4*)zero_me)[i] = make_float4(0.f, 0.f, 0.f, 0.f);
}

// ---------------------------------------------------------------------------
// Final fused stage. One wave32 handles 16 batch rows.
// gamma = sum(exp(u)*sigmoid(v)) / sum(exp(u))   over the 80-dim vectors
// (softmax max-shift cancels in the ratio).
// Row-sums via V_WMMA_F32_16X16X4_F32 with an all-ones B matrix:
//   D[m,n] = sum_k A[m,k] * 1, accumulated over 20 chunks of K=4.
// A-fragment layout (f32 16x4): lanes 0-15 hold M=0..15, K = 4c+{0,1};
//                               lanes 16-31 hold M=0..15, K = 4c+{2,3}.
// D layout: VGPR v -> M=v (lanes 0-15) / M=v+8 (lanes 16-31); all N equal.
// ---------------------------------------------------------------------------
__global__ void __launch_bounds__(256)
lg_gamma_kernel(const float* __restrict__ acc,
                const float* __restrict__ exu,
                const float* __restrict__ exi1,
                const float* __restrict__ exi0,
                const float* __restrict__ wu,
                const float* __restrict__ wi,
                const int* __restrict__ users,
                const int* __restrict__ items,
                const int* __restrict__ xij,
                float* __restrict__ gamma) {
    const int wave = (int)((blockIdx.x * blockDim.x + threadIdx.x) >> 5);
    const int lane = (int)(threadIdx.x & 31);
    const int half = lane >> 4;          // 0: K={0,1}, 1: K={2,3} of each chunk
    const int m    = lane & 15;          // row within this wave's 16-row tile
    const int row  = wave * 16 + m;      // BATCH = 8192 = 512 waves exactly

    const int u  = users[row];
    const int it = items[row];
    const int xf = xij[row];
    const float w_u = wu[0] * 0.25f;     // fold light_out = acc/(N_LAYERS+1)
    const float w_i = wi[0] * 0.25f;

    const float* ubase = acc + (long long)u * LATENT;
    const float* ibase = acc + (long long)(NUM_USERS + it) * LATENT;
    const float* uxij  = exu + (long long)u * XDIM;
    const float* ixij  = (xf ? exi1 : exi0) + (long long)it * XDIM;

    v8f c_es = {};                       // accumulates sum(e*s) per row
    v8f c_e  = {};                       // accumulates sum(e)   per row
    v2f ones; ones.x = 1.0f; ones.y = 1.0f;   // B = all-ones (4x16)

    for (int chunk = 0; chunk < 20; ++chunk) {   // 20 * K4 = 80 dims
        const int k0 = chunk * 4 + half * 2;
        float u0, u1, v0, v1;
        if (chunk < 16) {                // dims 0..63: scaled light_out
            u0 = w_u * ubase[k0];     u1 = w_u * ubase[k0 + 1];
            v0 = w_i * ibase[k0];     v1 = w_i * ibase[k0 + 1];
        } else {                         // dims 64..79: xij embeddings
            const int j = k0 - LATENT;
            u0 = uxij[j];             u1 = uxij[j + 1];
            v0 = ixij[j];             v1 = ixij[j + 1];
        }
        const float e0 = __expf(u0);
        const float e1 = __expf(u1);
        const float s0 = 1.0f / (1.0f + __expf(-v0));
        const float s1 = 1.0f / (1.0f + __expf(-v1));

        v2f a_es; a_es.x = e0 * s0; a_es.y = e1 * s1;
        v2f a_e;  a_e.x  = e0;      a_e.y  = e1;

        c_es = __builtin_amdgcn_wmma_f32_16x16x4_f32(
                   false, a_es, false, ones, (short)0, c_es, false, false);
        c_e  = __builtin_amdgcn_wmma_f32_16x16x4_f32(
                   false, a_e,  false, ones, (short)0, c_e,  false, false);
    }

    // lane 0 holds row-sums for rows 0..7 in c[0..7]; lane 16 for rows 8..15
    if (m == 0) {
        const int base = wave * 16 + half * 8;
        #pragma unroll
        for (int v = 0; v < 8; ++v)
            gamma[base + v] = c_es[v] / c_e[v];
    }
}

// ---------------------------------------------------------------------------
extern "C" void kernel_launch(void* const* d_in, const int* in_sizes, int n_in,
                              void* d_out, int out_size, void* d_ws, size_t ws_size,
                              hipStream_t stream) {
    const float* emb_user     = (const float*)d_in[0];
    const float* emb_item     = (const float*)d_in[1];
    const float* emb_user_xij = (const float*)d_in[2];
    const float* emb_item_x1  = (const float*)d_in[3];
    const float* emb_item_x0  = (const float*)d_in[4];
    const float* w_user       = (const float*)d_in[5];
    const float* w_item       = (const float*)d_in[6];
    const float* graph_vals   = (const float*)d_in[7];
    const int*   graph_rows   = (const int*)d_in[8];
    const int*   graph_cols   = (const int*)d_in[9];
    const int*   users        = (const int*)d_in[10];
    const int*   items        = (const int*)d_in[11];
    const int*   xij          = (const int*)d_in[12];
    float* gamma = (float*)d_out;

    const long long nelem = (long long)N_NODES * LATENT;   // 9.6M floats
    float* cur = (float*)d_ws;
    float* nxt = cur + nelem;
    float* acc = nxt + nelem;

    // init: cur = concat(emb), acc = cur, nxt = 0
    {
        int threads = 256;
        long long blocks = (nelem + threads - 1) / threads;
        lg_init_kernel<<<(unsigned)blocks, threads, 0, stream>>>(
            emb_user, emb_item, cur, acc, nxt);
    }

    // 3 LightGCN layers: spmm(cur -> nxt); acc += nxt; zero cur; swap
    for (int layer = 0; layer < N_LAYERS; ++layer) {
        {
            int threads = 256;
            long long work = NNZ_CNT * 16;
            long long blocks = (work + threads - 1) / threads;
            lg_spmm_kernel<<<(unsigned)blocks, threads, 0, stream>>>(
                graph_rows, graph_cols, graph_vals, cur, nxt);
        }
        {
            int threads = 256;
            long long n4 = nelem / 4;
            long long blocks = (n4 + threads - 1) / threads;
            lg_accum_kernel<<<(unsigned)blocks, threads, 0, stream>>>(
                acc, nxt, cur);
        }
        float* t = cur; cur = nxt; nxt = t;
    }

    // fused gather + softmax/sigmoid + dot via WMMA row-reductions
    {
        int threads = 256;                      // 8 waves/block, 16 rows/wave
        int blocks  = BATCH / (16 * 8);         // 8192 rows -> 64 blocks
        lg_gamma_kernel<<<blocks, threads, 0, stream>>>(
            acc, emb_user_xij, emb_item_x1, emb_item_x0,
            w_user, w_item, users, items, xij, gamma);
    }
}